// Pointnet2MSG_Feature_927712936448
// MI455X (gfx1250) — compile-verified
//
#include <hip/hip_runtime.h>
#include <hip/hip_bf16.h>

typedef __attribute__((ext_vector_type(2))) float v2f;
typedef __attribute__((ext_vector_type(8))) float v8f;

#define BN_EPS 1e-5f

// ---------------------------------------------------------------------------
// split pointcloud -> xyz0, feat0
// ---------------------------------------------------------------------------
__global__ void split_kernel(const float* __restrict__ pc,
                             float* __restrict__ xyz0,
                             float* __restrict__ feat0, int npts) {
  int total = npts * 3;
  for (int i = blockIdx.x * blockDim.x + threadIdx.x; i < total;
       i += gridDim.x * blockDim.x) {
    int p = i / 3, c = i % 3;
    xyz0[i]  = pc[p * 6 + c];
    feat0[i] = pc[p * 6 + 3 + c];
  }
}

__global__ void fill_zero_kernel(float* __restrict__ p, int n) {
  for (int i = blockIdx.x * blockDim.x + threadIdx.x; i < n;
       i += gridDim.x * blockDim.x)
    p[i] = 0.0f;
}

// Fragment-ready weight staging for V_WMMA_F32_16X16X4_F32:
//   Wq[((k0/4)*Np + col)*4 + dk] = W[k0+dk][col]   (zeros in pad region)
// so each lane's B fragment (elements (k0+2*hi, col),(k0+2*hi+1, col)) is one
// contiguous 8-byte load at a loop-invariant base + kq*Np*16 bytes.
__global__ void pad_weight_kernel(const float* __restrict__ W,
                                  float* __restrict__ Wq, int K, int N, int Kp,
                                  int Np) {
  int total = (Kp >> 2) * Np * 4;
  for (int i = blockIdx.x * blockDim.x + threadIdx.x; i < total;
       i += gridDim.x * blockDim.x) {
    int dk  = i & 3;
    int col = (i >> 2) % Np;
    int kq  = (i >> 2) / Np;
    int k   = kq * 4 + dk;
    Wq[i] = (k < K && col < N) ? W[k * N + col] : 0.0f;
  }
}

// ---------------------------------------------------------------------------
// Farthest point sampling: one block per batch, dists resident in LDS.
// Emits idx[s] = last BEFORE update (matches lax.scan semantics: idx[0]=0).
// blockDim.x must be a power of two.
// ---------------------------------------------------------------------------
__global__ void fps_kernel(const float* __restrict__ xyz, int* __restrict__ idx,
                           int N, int S) {
  extern __shared__ unsigned char smem_raw[];
  float* dists = (float*)smem_raw;                 // N floats
  float* rval  = dists + N;                        // blockDim floats
  int*   rind  = (int*)(rval + blockDim.x);        // blockDim ints
  __shared__ int s_last;

  int b = blockIdx.x, tid = threadIdx.x, bd = blockDim.x;
  const float* xb = xyz + (size_t)b * N * 3;

  for (int i = tid; i < N; i += bd) dists[i] = 1e10f;
  if (tid == 0) s_last = 0;
  __syncthreads();

  for (int s = 0; s < S; ++s) {
    int last = s_last;
    if (tid == 0) idx[b * S + s] = last;
    float px = xb[last * 3], py = xb[last * 3 + 1], pz = xb[last * 3 + 2];
    float best = -1.0f;
    int bi = 0;
    for (int i = tid; i < N; i += bd) {
      float dx = xb[i * 3] - px, dy = xb[i * 3 + 1] - py, dz = xb[i * 3 + 2] - pz;
      float d  = dx * dx + dy * dy + dz * dz;
      float nd = fminf(dists[i], d);
      dists[i] = nd;
      if (nd > best) { best = nd; bi = i; }
    }
    rval[tid] = best; rind[tid] = bi;
    __syncthreads();
    for (int off = bd >> 1; off > 0; off >>= 1) {
      if (tid < off) {
        float v2 = rval[tid + off]; int i2 = rind[tid + off];
        if (v2 > rval[tid] || (v2 == rval[tid] && i2 < rind[tid])) {
          rval[tid] = v2; rind[tid] = i2;
        }
      }
      __syncthreads();
    }
    if (tid == 0) s_last = rind[0];
    __syncthreads();
  }
}

// ---------------------------------------------------------------------------
// Gather sampled centers: new_xyz[b,s,:] = xyz[b, idx[b,s], :]
// ---------------------------------------------------------------------------
__global__ void gather_centers_kernel(const float* __restrict__ xyz,
                                      const int* __restrict__ idx,
                                      float* __restrict__ new_xyz, int BS, int N,
                                      int S) {
  int total = BS * 3;
  for (int i = blockIdx.x * blockDim.x + threadIdx.x; i < total;
       i += gridDim.x * blockDim.x) {
    int bs = i / 3, c = i % 3;
    int b = bs / S;
    int id = idx[bs];
    new_xyz[i] = xyz[((size_t)b * N + id) * 3 + c];
  }
}

// ---------------------------------------------------------------------------
// Fused ball query + gather + relative-xyz concat, padded row stride Cp with
// zero fill in [C,Cp). One wave (32 lanes) per center (wave32 ballot/popc).
// Selection = first nsample indices within radius in ascending index order,
// padded with first hit (or clamped N-1 if no hit, matching JAX OOB clamp).
// ---------------------------------------------------------------------------
__global__ void ballquery_gather_kernel(const float* __restrict__ xyz,
                                        const float* __restrict__ feats,
                                        const float* __restrict__ new_xyz,
                                        float* __restrict__ h0, int B, int N,
                                        int S, int Cfeat, int Cp, int nsample,
                                        float r2) {
  __shared__ int sidx[8][32];
  int widx = threadIdx.x >> 5, lane = threadIdx.x & 31;
  int wg = blockIdx.x * 8 + widx;
  if (wg >= B * S) return;
  int b = wg / S;
  float cx = new_xyz[wg * 3], cy = new_xyz[wg * 3 + 1], cz = new_xyz[wg * 3 + 2];
  const float* xb = xyz + (size_t)b * N * 3;

  int count = 0;
  for (int base = 0; base < N && count < nsample; base += 32) {
    int i = base + lane;
    float d2 = 1e30f;
    if (i < N) {
      float dx = xb[i * 3] - cx, dy = xb[i * 3 + 1] - cy, dz = xb[i * 3 + 2] - cz;
      d2 = dx * dx + dy * dy + dz * dz;
    }
    bool p = d2 < r2;
    unsigned mask = (unsigned)__ballot(p);   // wave32: low 32 bits exact
    int pos = count + __popc(mask & ((1u << lane) - 1u));
    if (p && pos < nsample) sidx[widx][pos] = i;
    count += __popc(mask);
  }
  if (count > nsample) count = nsample;
  int pad = (count == 0) ? (N - 1) : -1;
  for (int j = count + lane; j < nsample; j += 32) sidx[widx][j] = pad;

  int C = Cfeat + 3;
  float* out = h0 + (size_t)wg * nsample * Cp;
  const float* fb = feats + (size_t)b * N * Cfeat;
  for (int j = 0; j < nsample; ++j) {
    int id = sidx[widx][j];
    if (id < 0) id = sidx[widx][0];
    for (int c = lane; c < Cp; c += 32) {
      float v = 0.0f;
      if (c == 0)      v = xb[id * 3 + 0] - cx;
      else if (c == 1) v = xb[id * 3 + 1] - cy;
      else if (c == 2) v = xb[id * 3 + 2] - cz;
      else if (c < C)  v = fb[(size_t)id * Cfeat + (c - 3)];
      out[(size_t)j * Cp + c] = v;
    }
  }
}

// ---------------------------------------------------------------------------
// f32 GEMM on the matrix pipe: C[M,Np] = A[M,Kp] * W[Kp,Np] via
// V_WMMA_F32_16X16X4_F32. All dims tile-padded (Kp%16==0, Np%16==0, M%16==0,
// pad regions zero) and weights pre-staged fragment-ready, so the inner loop
// is exactly: one b64 A load + one b64 B load + one WMMA per K-quad,
// with loop-invariant bases and constant-stride offsets.
// A-frag layout: lanes0-15 row M=l K={k,k+1}; lanes16-31 row M=l K={k+2,k+3}.
// C/D layout: VGPR r -> (M = tm + r + 8*hi, N = tn + l).
// ---------------------------------------------------------------------------
__global__ void gemm_wmma_kernel(const float* __restrict__ A,
                                 const float* __restrict__ Wq,
                                 float* __restrict__ C, int M, int Kp, int Np) {
  int wave = (blockIdx.x * blockDim.x + threadIdx.x) >> 5;
  int lane = threadIdx.x & 31;
  int tiles_n = Np >> 4;
  int tiles = (M >> 4) * tiles_n;
  if (wave >= tiles) return;
  int tm = (wave / tiles_n) << 4;
  int tn = (wave % tiles_n) << 4;
  int l = lane & 15, hi = lane >> 4;

  const float* aptr = A + (size_t)(tm + l) * Kp + 2 * hi;          // 8B aligned
  const float* bptr = Wq + ((size_t)(tn + l) << 2) + 2 * hi;       // 8B aligned
  size_t bstride = (size_t)Np << 2;  // floats per K-quad in staged layout

  v8f acc = {};
#pragma unroll 4
  for (int q = 0; q < (Kp >> 2); ++q) {
    v2f a   = *(const v2f*)(aptr + (size_t)q * 4);
    v2f bfr = *(const v2f*)(bptr + (size_t)q * bstride);
    acc = __builtin_amdgcn_wmma_f32_16x16x4_f32(false, a, false, bfr, (short)0,
                                                acc, false, false);
  }

  float* cptr = C + (size_t)(tm + (hi << 3)) * Np + tn + l;
#pragma unroll
  for (int r = 0; r < 8; ++r) cptr[(size_t)r * Np] = acc[r];
}

// ---------------------------------------------------------------------------
// Per-channel sum / sum-of-squares (BN statistics), LDS pre-accumulation.
// Input has padded stride Cp; pad columns skipped.
// ---------------------------------------------------------------------------
__global__ void stats_kernel(const float* __restrict__ h, int M, int C, int Cp,
                             float* __restrict__ sums,
                             float* __restrict__ sumsq) {
  __shared__ float ss[512];
  __shared__ float sq[512];
  for (int c = threadIdx.x; c < C; c += blockDim.x) { ss[c] = 0; sq[c] = 0; }
  __syncthreads();
  size_t total = (size_t)M * Cp;
  for (size_t i = (size_t)blockIdx.x * blockDim.x + threadIdx.x; i < total;
       i += (size_t)gridDim.x * blockDim.x) {
    int c = (int)(i % Cp);
    if (c < C) {
      float v = h[i];
      atomicAdd(&ss[c], v);
      atomicAdd(&sq[c], v * v);
    }
  }
  __syncthreads();
  for (int c = threadIdx.x; c < C; c += blockDim.x) {
    atomicAdd(&sums[c], ss[c]);
    atomicAdd(&sumsq[c], sq[c]);
  }
}

// BN + ReLU in place; pad columns forced to exact zero so the buffer can be
// consumed as a zero-K-padded GEMM A operand.
__global__ void bn_relu_kernel(float* __restrict__ h, int M, int C, int Cp,
                               const float* __restrict__ sums,
                               const float* __restrict__ sumsq,
                               const float* __restrict__ gamma,
                               const float* __restrict__ beta) {
  size_t total = (size_t)M * Cp;
  float inv = 1.0f / (float)M;
  for (size_t i = (size_t)blockIdx.x * blockDim.x + threadIdx.x; i < total;
       i += (size_t)gridDim.x * blockDim.x) {
    int c = (int)(i % Cp);
    float outv = 0.0f;
    if (c < C) {
      float mu = sums[c] * inv;
      float var = sumsq[c] * inv - mu * mu;
      float v = (h[i] - mu) * rsqrtf(var + BN_EPS) * gamma[c] + beta[c];
      outv = v > 0.0f ? v : 0.0f;
    }
    h[i] = outv;
  }
}

// ---------------------------------------------------------------------------
// Max-pool over nsample axis (input stride Cp), writing into channel slice
// [coff, coff+C) of l_feat[k+1] (row stride Ctot).
// ---------------------------------------------------------------------------
__global__ void maxpool_kernel(const float* __restrict__ h,
                               float* __restrict__ out, int rows, int nsample,
                               int C, int Cp, int Ctot, int coff) {
  int total = rows * C;
  for (int i = blockIdx.x * blockDim.x + threadIdx.x; i < total;
       i += gridDim.x * blockDim.x) {
    int bs = i / C, c = i % C;
    const float* p = h + ((size_t)bs * nsample) * Cp + c;
    float m = p[0];
    for (int j = 1; j < nsample; ++j) m = fmaxf(m, p[(size_t)j * Cp]);
    out[(size_t)bs * Ctot + coff + c] = m;
  }
}

// ---------------------------------------------------------------------------
// Host orchestration
// ---------------------------------------------------------------------------
static inline int iceil(int a, int b) { return (a + b - 1) / b; }
static inline int pad16(int x) { return (x + 15) & ~15; }

extern "C" void kernel_launch(void* const* d_in, const int* in_sizes, int n_in,
                              void* d_out, int out_size, void* d_ws,
                              size_t ws_size, hipStream_t stream) {
  (void)in_sizes; (void)n_in; (void)out_size; (void)ws_size;

  static const float RADII[4][2] = {{0.05f, 0.1f}, {0.1f, 0.2f},
                                    {0.2f, 0.4f},  {0.4f, 0.8f}};
  static const int NSAMP[2] = {16, 32};
  static const int MLPS[4][2][3] = {
      {{16, 16, 32}, {32, 32, 64}},
      {{64, 64, 128}, {64, 96, 128}},
      {{128, 196, 256}, {128, 196, 256}},
      {{256, 256, 512}, {256, 384, 512}}};
  const int B = 4;
  const int Ns[5] = {16384, 1024, 256, 64, 16};
  const int Cf[5] = {3, 96, 256, 512, 1024};

  float* out = (float*)d_out;
  // Output layout: l_xyz[0..4] then l_feat[0..4], flat-concatenated.
  size_t xoff[5], foff[5];
  xoff[0] = 0;
  for (int k = 1; k < 5; ++k) xoff[k] = xoff[k - 1] + (size_t)B * Ns[k - 1] * 3;
  foff[0] = xoff[4] + (size_t)B * Ns[4] * 3;
  for (int k = 1; k < 5; ++k)
    foff[k] = foff[k - 1] + (size_t)B * Ns[k - 1] * Cf[k - 1];

  // Workspace: two ping-pong activation buffers (tile-padded strides) +
  // BN stats + fragment-staged weights + FPS indices.
  const size_t HBUF = 8u * 1024u * 1024u + 65536u;  // floats; max 131072*64
  float* hA    = (float*)d_ws;
  float* hB    = hA + HBUF;
  float* sums  = hB + HBUF;
  float* sumsq = sums + 512;
  float* Wq    = sumsq + 512;           // up to 528*512 floats
  int*   fidx  = (int*)(Wq + 528 * 512);

  // split pointcloud
  {
    int total = B * Ns[0] * 3;
    split_kernel<<<iceil(total, 256), 256, 0, stream>>>(
        (const float*)d_in[0], out + xoff[0], out + foff[0], B * Ns[0]);
  }

  int pi = 1;  // param pointer cursor: (W, gamma, beta) per layer
  const int fps_block[4] = {1024, 512, 128, 64};

  for (int k = 0; k < 4; ++k) {
    int N = Ns[k], S = Ns[k + 1];
    const float* xyz_in  = out + xoff[k];
    const float* feat_in = out + foff[k];
    float* xyz_out = out + xoff[k + 1];
    float* feat_out = out + foff[k + 1];

    // FPS
    {
      int bd = fps_block[k];
      size_t shmem = (size_t)N * 4 + (size_t)bd * 8;
      fps_kernel<<<B, bd, shmem, stream>>>(xyz_in, fidx, N, S);
    }
    // gather centers
    gather_centers_kernel<<<iceil(B * S * 3, 256), 256, 0, stream>>>(
        xyz_in, fidx, xyz_out, B * S, N, S);

    int coff = 0;
    for (int br = 0; br < 2; ++br) {
      int nsample = NSAMP[br];
      float r = RADII[k][br];
      int Cin = Cf[k] + 3;
      int CinP = pad16(Cin);
      int M = B * S * nsample;

      // ball query + gather into hA (padded, zero-filled pad columns)
      {
        int waves = B * S;
        ballquery_gather_kernel<<<iceil(waves, 8), 256, 0, stream>>>(
            xyz_in, feat_in, xyz_out, hA, B, N, S, Cf[k], CinP, nsample, r * r);
      }

      float* cur = hA;
      float* nxt = hB;
      int C_in = Cin, C_inP = CinP;
      for (int layer = 0; layer < 3; ++layer) {
        int Cout = MLPS[k][br][layer];
        int CoutP = pad16(Cout);
        const float* W     = (const float*)d_in[pi + 0];
        const float* gamma = (const float*)d_in[pi + 1];
        const float* beta  = (const float*)d_in[pi + 2];
        pi += 3;

        // stage fragment-ready zero-padded weights + zero BN stats
        pad_weight_kernel<<<iceil(C_inP * CoutP, 256), 256, 0, stream>>>(
            W, Wq, C_in, Cout, C_inP, CoutP);
        fill_zero_kernel<<<iceil(Cout, 256), 256, 0, stream>>>(sums, Cout);
        fill_zero_kernel<<<iceil(Cout, 256), 256, 0, stream>>>(sumsq, Cout);

        // WMMA GEMM (guard-free inner loop)
        {
          int tiles = (M >> 4) * (CoutP >> 4);
          int threads = tiles * 32;
          gemm_wmma_kernel<<<iceil(threads, 256), 256, 0, stream>>>(
              cur, Wq, nxt, M, C_inP, CoutP);
        }
        // BN stats + normalize + relu (in place on nxt)
        {
          size_t total = (size_t)M * CoutP;
          int grid = (int)((total + 255) / 256);
          if (grid > 2048) grid = 2048;
          stats_kernel<<<grid, 256, 0, stream>>>(nxt, M, Cout, CoutP, sums,
                                                 sumsq);
          bn_relu_kernel<<<grid, 256, 0, stream>>>(nxt, M, Cout, CoutP, sums,
                                                   sumsq, gamma, beta);
        }
        float* t = cur; cur = nxt; nxt = t;
        C_in = Cout; C_inP = CoutP;
      }

      // max-pool over nsample into feat_out channel slice
      {
        int total = B * S * C_in;
        maxpool_kernel<<<iceil(total, 256), 256, 0, stream>>>(
            cur, feat_out, B * S, nsample, C_in, C_inP, Cf[k + 1], coff);
      }
      coff += C_in;
    }
  }
}